// HungarianMatcher_50379966382508
// MI455X (gfx1250) — compile-verified
//
#include <hip/hip_runtime.h>
#include <hip/hip_bf16.h>

// ---------------------------------------------------------------------------
// HungarianMatcher for MI455X (gfx1250, wave32).
//
// Kernel 1 (cost_kernel): one workgroup (256 thr = 8 waves) per generator row
//   r in [0, 8192). TDM-stages the 8 big logit rows (8 KB each) into LDS,
//   computes softmax stats (max, sum-exp) per tensor with wave32 butterflies,
//   overwrites LDS with exp values, then computes
//       cost[r][j] = -sum_t inv_denom_t * E_t[ gold_t[j] ]
//   using chained V_WMMA_F32_16X16X4_F32 (16 columns per WMMA group,
//   k-slots = tensors, B = broadcast inv_denom weights).
//   HBM traffic ~570 MB -> ~25 us at 23.3 TB/s (memory-bound by design).
//
// Kernel 2 (lsa_kernel): 64 blocks x 1 wave; exact Jonker-Volgenant shortest
//   augmenting path on the transposed 16x128 slice, lane-parallel over the
//   128 columns, matching scipy/reference tie-breaking (first argmin).
// ---------------------------------------------------------------------------

#define BSZ   64
#define NGEN  128
#define NCLS  30
#define SEQ   2048
#define G     16
#define NGOLD (BSZ * G)          // 1024
#define NROWS (BSZ * NGEN)      // 8192

typedef float v2f  __attribute__((ext_vector_type(2)));
typedef float v8f  __attribute__((ext_vector_type(8)));
typedef unsigned int v4u __attribute__((ext_vector_type(4)));
typedef int   v8i  __attribute__((ext_vector_type(8)));
typedef int   v4i  __attribute__((ext_vector_type(4)));

#define LOG2E 1.44269504088896340736f

// LDS layout (dynamic):
//   [0,      65536)  : Ebuf, 8 tensors x 2048 f32 (exp values after pass 2)
//   [65536,  65664)  : relE, 32 f32 (exp values of rel row)
//   [65664,  65792)  : stats, 9 f32 inv_denom (+pad)
//   [65792, 102656)  : gidx, 9 x 1024 i32 gold indices
#define LDS_RELE  65536
#define LDS_STATS 65664
#define LDS_GIDX  65792
#define LDS_TOTAL 102656

// Issue one TDM descriptor: load `nelem` f32 (one row) from global -> LDS.
// D# group0: count=1, lds_addr, global_addr[56:0], type=2.
// D# group1: data_size=4B, tensor_dim0=nelem, tensor_dim1=1,
//            tile_dim0=nelem, tile_dim1=1, tensor_dim0_stride=nelem.
// Groups 2/3 (and the trailing group) are zero: 1D/2D tensor, no gather,
// no iteration.  6-arg builtin form (clang-23 / therock-10.0 headers).
__device__ __forceinline__ void tdm_load_row(unsigned long long gaddr,
                                             unsigned int lds_off,
                                             unsigned int nelem) {
  v4u g0;
  g0.x = 1u;                                            // count=1, user mode
  g0.y = lds_off;                                       // lds_addr (bytes)
  g0.z = (unsigned int)(gaddr & 0xFFFFFFFFull);         // global_addr[31:0]
  g0.w = (unsigned int)((gaddr >> 32) & 0x01FFFFFFull)  // global_addr[56:32]
         | (2u << 30);                                  // type=2 ("image")
  v8i g1;
  g1[0] = (int)(2u << 16);            // workgroup_mask=0, data_size=2 (4B)
  g1[1] = (int)(nelem << 16);         // tensor_dim0[15:0] @ bits 63:48
  g1[2] = (int)((nelem >> 16) | (1u << 16)); // dim0[31:16] + tensor_dim1=1
  g1[3] = (int)(nelem << 16);         // tile_dim0 @ bits 127:112
  g1[4] = 1;                          // tile_dim1=1, tile_dim2=0
  g1[5] = (int)nelem;                 // tensor_dim0_stride[31:0]
  g1[6] = 0;                          // stride hi / dim1_stride lo
  g1[7] = 0;
  v4i z4 = {0, 0, 0, 0};
  v8i z8 = {0, 0, 0, 0, 0, 0, 0, 0};
  __builtin_amdgcn_tensor_load_to_lds(g0, g1, z4, z4, z8, 0);
}

__global__ __launch_bounds__(256)
void hm_cost_kernel(const float* __restrict__ rel,
                    const float* __restrict__ p1, const float* __restrict__ p2,
                    const float* __restrict__ p3, const float* __restrict__ p4,
                    const float* __restrict__ p5, const float* __restrict__ p6,
                    const float* __restrict__ p7, const float* __restrict__ p8,
                    const int* __restrict__ g0, const int* __restrict__ g1,
                    const int* __restrict__ g2, const int* __restrict__ g3,
                    const int* __restrict__ g4, const int* __restrict__ g5,
                    const int* __restrict__ g6, const int* __restrict__ g7,
                    const int* __restrict__ g8,
                    float* __restrict__ out) {
  extern __shared__ unsigned char smem[];
  float* Ebuf  = (float*)smem;
  float* relE  = (float*)(smem + LDS_RELE);
  float* stats = (float*)(smem + LDS_STATS);
  int*   gidx  = (int*)(smem + LDS_GIDX);

  const int tid  = threadIdx.x;
  const int lane = tid & 31;
  const int wid  = tid >> 5;
  const int r    = blockIdx.x;

  const float* pos[8] = {p1, p2, p3, p4, p5, p6, p7, p8};
  const int*  gold[9] = {g0, g1, g2, g3, g4, g5, g6, g7, g8};

  if (wid == 0) {
    // --- Tensor Data Mover: 8 async row loads into LDS (wave 0 issues). ---
    unsigned int ebase = (unsigned int)(size_t)Ebuf;  // low 32 bits = LDS off
#pragma unroll
    for (int t = 0; t < 8; ++t) {
      unsigned long long ga =
          (unsigned long long)(size_t)(pos[t] + (size_t)r * SEQ);
      tdm_load_row(ga, ebase + (unsigned int)t * (SEQ * 4), SEQ);
    }
    // --- rel (width 30) softmax, done by wave 0 while TDM runs. ---
    float x = (lane < NCLS) ? rel[(size_t)r * NCLS + lane] : -3.0e38f;
    float m = x;
#pragma unroll
    for (int d = 16; d; d >>= 1) m = fmaxf(m, __shfl_xor(m, d, 32));
    float e = (lane < NCLS) ? __builtin_amdgcn_exp2f((x - m) * LOG2E) : 0.0f;
    relE[lane] = e;
    float s = e;
#pragma unroll
    for (int d = 16; d; d >>= 1) s += __shfl_xor(s, d, 32);
    if (lane == 0) stats[0] = 1.0f / s;
    __builtin_amdgcn_s_wait_tensorcnt(0);
  } else {
    // --- Waves 1..7: stage the 9x1024 gold index arrays into LDS. ---
#pragma unroll
    for (int t = 0; t < 9; ++t) {
      const int* gp = gold[t];
      for (int j = tid - 32; j < NGOLD; j += 224) gidx[t * NGOLD + j] = gp[j];
    }
  }
  __syncthreads();

  // --- Per-tensor softmax stats; overwrite LDS row with exp values. ---
  {
    float* row = Ebuf + wid * SEQ;
    float m = -3.0e38f;
    for (int k = lane; k < SEQ; k += 32) m = fmaxf(m, row[k]);
#pragma unroll
    for (int d = 16; d; d >>= 1) m = fmaxf(m, __shfl_xor(m, d, 32));
    float s = 0.0f;
    for (int k = lane; k < SEQ; k += 32) {
      float e = __builtin_amdgcn_exp2f((row[k] - m) * LOG2E);
      row[k] = e;
      s += e;
    }
#pragma unroll
    for (int d = 16; d; d >>= 1) s += __shfl_xor(s, d, 32);
    if (lane == 0) stats[1 + wid] = 1.0f / s;
  }
  __syncthreads();

  // --- Gather + WMMA accumulate: 16 columns per group, 3 chained WMMAs. ---
  // A[m][k] = E_t[gold_t[jbase+m]], B[k][n] = inv_denom_t (bcast over n),
  // D[m][*] = sum_t inv_denom_t * E_t[...]  -> cost = -D.
  // f32 16x16x4 A layout: lane = m + 16*(k>=2), vgpr = k&1. B mirrored.
  const int h    = lane >> 4;    // k half: 0 -> k in {0,1}, 1 -> k in {2,3}
  const int mcol = lane & 15;    // column within group (M index)
  for (int grp = wid; grp < NGOLD / 16; grp += 8) {
    const int jbase = grp * 16;
    const int j     = jbase + mcol;
    v8f c = {0.f, 0.f, 0.f, 0.f, 0.f, 0.f, 0.f, 0.f};
    {  // tensors 1..4
      const int ta = 1 + 2 * h, tb = 2 + 2 * h;
      v2f a, b;
      a.x = Ebuf[(ta - 1) * SEQ + gidx[ta * NGOLD + j]];
      a.y = Ebuf[(tb - 1) * SEQ + gidx[tb * NGOLD + j]];
      b.x = stats[ta];
      b.y = stats[tb];
      c = __builtin_amdgcn_wmma_f32_16x16x4_f32(false, a, false, b,
                                                (short)0, c, false, false);
    }
    {  // tensors 5..8
      const int ta = 5 + 2 * h, tb = 6 + 2 * h;
      v2f a, b;
      a.x = Ebuf[(ta - 1) * SEQ + gidx[ta * NGOLD + j]];
      a.y = Ebuf[(tb - 1) * SEQ + gidx[tb * NGOLD + j]];
      b.x = stats[ta];
      b.y = stats[tb];
      c = __builtin_amdgcn_wmma_f32_16x16x4_f32(false, a, false, b,
                                                (short)0, c, false, false);
    }
    {  // tensor 0 (rel) in k-slot 0, other slots zero
      v2f a = {0.f, 0.f}, b = {0.f, 0.f};
      if (h == 0) {
        a.x = relE[gidx[j]];
        b.x = stats[0];
      }
      c = __builtin_amdgcn_wmma_f32_16x16x4_f32(false, a, false, b,
                                                (short)0, c, false, false);
    }
    // D rows constant over N: lane 0 holds M=0..7, lane 16 holds M=8..15.
    if (mcol == 0) {
      const size_t off = (size_t)r * NGOLD + (size_t)jbase + (size_t)h * 8;
#pragma unroll
      for (int v = 0; v < 8; ++v) out[off + v] = -c[v];
    }
  }
}

// ---------------------------------------------------------------------------
// Exact JV shortest-augmenting-path LSA on the transposed 16x128 slice.
// One wave per batch element; lanes own columns {lane, lane+32, ...}.
// ---------------------------------------------------------------------------
__global__ __launch_bounds__(32)
void hm_lsa_kernel(const float* __restrict__ cost,
                   float* __restrict__ rows_out,
                   float* __restrict__ cols_out) {
  const int b    = blockIdx.x;
  const int lane = threadIdx.x;

  __shared__ float C[G * NGEN];     // Ct[i][j] = cost[b, j, b*16+i]
  __shared__ float u[G + 1];
  __shared__ float vv[NGEN + 1];
  __shared__ float minv[NGEN + 1];
  __shared__ int   p[NGEN + 1];
  __shared__ int   way[NGEN + 1];
  __shared__ int   used[NGEN + 1];
  __shared__ int   j0s;
  __shared__ int   colarr[G];

  for (int idx = lane; idx < G * NGEN; idx += 32) {
    const int i = idx >> 7, j = idx & (NGEN - 1);
    C[idx] = cost[((size_t)b * NGEN + j) * NGOLD + (size_t)b * G + i];
  }
  for (int idx = lane; idx <= NGEN; idx += 32) {
    vv[idx] = 0.0f; p[idx] = 0; way[idx] = 0;
  }
  if (lane <= G) u[lane] = 0.0f;
  __syncthreads();

  const float INF = __builtin_huge_valf();
  for (int i = 1; i <= G; ++i) {
    for (int idx = lane; idx <= NGEN; idx += 32) { minv[idx] = INF; used[idx] = 0; }
    if (lane == 0) { p[0] = i; j0s = 0; }
    __syncthreads();
    while (true) {
      const int j0 = j0s;
      if (lane == 0) used[j0] = 1;
      __syncthreads();
      const int   i0  = p[j0];
      const float ui0 = u[i0];
      float bestv = INF;
      int   bestj = NGEN + 1;
#pragma unroll
      for (int q = 0; q < 4; ++q) {
        const int j = 1 + lane + 32 * q;
        if (!used[j]) {
          const float cur = C[(i0 - 1) * NGEN + (j - 1)] - ui0 - vv[j];
          if (cur < minv[j]) { minv[j] = cur; way[j] = j0; }
          const float mv = minv[j];
          if (mv < bestv) { bestv = mv; bestj = j; }  // first-min within lane
        }
      }
      // lexicographic (value, index) wave argmin -> matches np.argmin ties
#pragma unroll
      for (int d = 16; d; d >>= 1) {
        const float ov = __shfl_xor(bestv, d, 32);
        const int   oj = __shfl_xor(bestj, d, 32);
        if (ov < bestv || (ov == bestv && oj < bestj)) { bestv = ov; bestj = oj; }
      }
      const float delta = bestv;
      const int   j1    = bestj;
      __syncthreads();
#pragma unroll
      for (int q = 0; q < 5; ++q) {
        const int j = lane + 32 * q;
        if (j <= NGEN) {
          if (used[j]) { u[p[j]] += delta; vv[j] -= delta; }
          else         { minv[j] -= delta; }
        }
      }
      if (lane == 0) j0s = j1;
      __syncthreads();
      if (p[j1] == 0) break;
    }
    if (lane == 0) {  // augment along `way`
      int j0 = j0s;
      while (j0) { const int j1 = way[j0]; p[j0] = p[j1]; j0 = j1; }
    }
    __syncthreads();
  }

  if (lane == 0) {
    for (int j = 1; j <= NGEN; ++j)
      if (p[j]) colarr[p[j] - 1] = j - 1;
    int ord[G];
    for (int k = 0; k < G; ++k) ord[k] = k;
    for (int a = 0; a < G; ++a) {        // selection sort by col value
      int best = a;
      for (int q = a + 1; q < G; ++q)
        if (colarr[ord[q]] < colarr[ord[best]]) best = q;
      const int t = ord[a]; ord[a] = ord[best]; ord[best] = t;
      rows_out[b * G + a] = (float)colarr[ord[a]];  // sorted generator idx
      cols_out[b * G + a] = (float)ord[a];          // matching gold idx
    }
  }
}

extern "C" void kernel_launch(void* const* d_in, const int* in_sizes, int n_in,
                              void* d_out, int out_size, void* d_ws, size_t ws_size,
                              hipStream_t stream) {
  const float* rel = (const float*)d_in[0];
  const float* p1  = (const float*)d_in[1];
  const float* p2  = (const float*)d_in[2];
  const float* p3  = (const float*)d_in[3];
  const float* p4  = (const float*)d_in[4];
  const float* p5  = (const float*)d_in[5];
  const float* p6  = (const float*)d_in[6];
  const float* p7  = (const float*)d_in[7];
  const float* p8  = (const float*)d_in[8];
  const int* g0 = (const int*)d_in[9];
  const int* g1 = (const int*)d_in[10];
  const int* g2 = (const int*)d_in[11];
  const int* g3 = (const int*)d_in[12];
  const int* g4 = (const int*)d_in[13];
  const int* g5 = (const int*)d_in[14];
  const int* g6 = (const int*)d_in[15];
  const int* g7 = (const int*)d_in[16];
  const int* g8 = (const int*)d_in[17];

  float* out_cost = (float*)d_out;
  const size_t cost_elems = (size_t)NROWS * NGOLD;  // 8,388,608
  float* out_rows = out_cost + cost_elems;
  float* out_cols = out_rows + (size_t)BSZ * G;

  hm_cost_kernel<<<NROWS, 256, LDS_TOTAL, stream>>>(
      rel, p1, p2, p3, p4, p5, p6, p7, p8,
      g0, g1, g2, g3, g4, g5, g6, g7, g8, out_cost);
  hm_lsa_kernel<<<BSZ, 32, 0, stream>>>(out_cost, out_rows, out_cols);
}